// GNN_37941741093521
// MI455X (gfx1250) — compile-verified
//
#include <hip/hip_runtime.h>
#include <math.h>
#include <stdint.h>

typedef float v2f __attribute__((ext_vector_type(2)));
typedef float v8f __attribute__((ext_vector_type(8)));

#define N_NODES   100000
#define N_FEAT    1000
#define HIDDEN    16
#define N_CLASSES 20
#define N_PAIRS   (N_NODES / 32)   // 3125 (2 x 16-row tiles per wave)

// ---------------------------------------------------------------- utilities
__global__ void init_f32(float* __restrict__ p, float v, int n) {
  int i = blockIdx.x * blockDim.x + threadIdx.x;
  if (i < n) p[i] = v;
}

__global__ void degree_kernel(const int* __restrict__ dst, int E,
                              float* __restrict__ deg) {
  int e = blockIdx.x * blockDim.x + threadIdx.x;
  if (e < E) atomicAdd(&deg[dst[e]], 1.0f);
}

__global__ void rsqrt_kernel(float* __restrict__ deg, int n) {
  int i = blockIdx.x * blockDim.x + threadIdx.x;
  if (i < n) deg[i] = rsqrtf(deg[i]);   // deg >= 1 always (self loop)
}

// ------------------------------------------------- GEMM1: h1 = x @ W1 (WMMA)
// 4 waves/block; each wave computes TWO 16x16 f32 output tiles (32 rows),
// reusing each B fragment for 2 WMMAs. W1 is DMA'd to LDS with
// global_load_async_to_lds_b128 (ASYNCcnt path).
__global__ void __launch_bounds__(128)
gemm1_wmma(const float* __restrict__ x, const float* __restrict__ W1,
           float* __restrict__ h1) {
  __shared__ __align__(16) float w1s[N_FEAT * HIDDEN];   // 64000 B

  // async preload W1 -> LDS in 16B chunks (memory -> LDS, no VGPR round-trip)
  {
    const char* gsrc = (const char*)W1;
    char* lbase = (char*)w1s;
    for (int byteoff = threadIdx.x * 16; byteoff < N_FEAT * HIDDEN * 4;
         byteoff += 128 * 16) {
      uint32_t laddr = (uint32_t)(uintptr_t)(lbase + byteoff);  // LDS offset
      const void* gaddr = (const void*)(gsrc + byteoff);
      asm volatile("global_load_async_to_lds_b128 %0, %1, off"
                   :: "v"(laddr), "v"(gaddr) : "memory");
    }
    asm volatile("s_wait_asynccnt 0x0" ::: "memory");  // this wave's DMAs done
  }
  __syncthreads();

  const int wave = threadIdx.x >> 5;                // wave32
  const int lane = threadIdx.x & 31;
  const int pair = blockIdx.x * 4 + wave;           // rows [pair*32, pair*32+32)
  if (pair >= N_PAIRS) return;                      // whole-wave uniform exit

  const int mrow = lane & 15;                       // A: M = lane % 16
  const int koff = (lane >> 4) << 1;                // lanes 16..31 hold K+2,K+3
  const int col  = lane & 15;                       // B: N = lane % 16
  const float* xrow0 = x + (size_t)(pair * 32 + mrow) * N_FEAT + koff;
  const float* xrow1 = xrow0 + (size_t)16 * N_FEAT;

  v8f acc0 = {0.f, 0.f, 0.f, 0.f, 0.f, 0.f, 0.f, 0.f};
  v8f acc1 = {0.f, 0.f, 0.f, 0.f, 0.f, 0.f, 0.f, 0.f};
  for (int k = 0; k < N_FEAT; k += 4) {
    v2f a0 = *(const v2f*)(xrow0 + k);              // A[M][k+koff .. +1]
    v2f a1 = *(const v2f*)(xrow1 + k);
    v2f b;                                          // B[k+koff .. +1][col]
    b.x = w1s[(k + koff) * HIDDEN + col];
    b.y = w1s[(k + koff + 1) * HIDDEN + col];
    acc0 = __builtin_amdgcn_wmma_f32_16x16x4_f32(
        false, a0, false, b, (short)0, acc0, false, false);
    acc1 = __builtin_amdgcn_wmma_f32_16x16x4_f32(
        false, a1, false, b, (short)0, acc1, false, false);
  }

  // C/D layout: VGPR r -> M = r (lanes 0-15) / r+8 (lanes 16-31), N = lane%16
  const int mbase = pair * 32 + ((lane >> 4) << 3);
  const int n = lane & 15;
#pragma unroll
  for (int r = 0; r < 8; ++r) {
    h1[(size_t)(mbase + r) * HIDDEN + n]      = acc0[r];
    h1[(size_t)(mbase + 16 + r) * HIDDEN + n] = acc1[r];
  }
}

// ------------------------------------------------- layer-1 message passing
__global__ void mp1_kernel(const int* __restrict__ src, const int* __restrict__ dst,
                           const float* __restrict__ dinv,
                           const float* __restrict__ h1,
                           float* __restrict__ agg1, int E) {
  int t = blockIdx.x * blockDim.x + threadIdx.x;
  if (t >= E * HIDDEN) return;
  int e = t >> 4;
  int j = t & 15;
  int s = src[e], d = dst[e];
  float w = dinv[s] * dinv[d];
  atomicAdd(&agg1[(size_t)d * HIDDEN + j], h1[(size_t)s * HIDDEN + j] * w);
}

// self-loop + bias + ReLU, in place (agg1 becomes layer-2 activation)
__global__ void finalize1(float* __restrict__ agg1, const float* __restrict__ h1,
                          const float* __restrict__ dinv,
                          const float* __restrict__ b1) {
  int t = blockIdx.x * blockDim.x + threadIdx.x;
  if (t >= N_NODES * HIDDEN) return;
  int i = t >> 4, j = t & 15;
  float di = dinv[i];
  float v = agg1[t] + h1[t] * di * di + b1[j];
  agg1[t] = v > 0.f ? v : 0.f;
}

// ------------------------------------------------- GEMM2: z = hact @ W2
__global__ void gemm2_kernel(const float* __restrict__ hact,
                             const float* __restrict__ W2,
                             float* __restrict__ z) {
  int t = blockIdx.x * blockDim.x + threadIdx.x;
  if (t >= N_NODES * N_CLASSES) return;
  int i = t / N_CLASSES, c = t - i * N_CLASSES;
  const float* hr = hact + (size_t)i * HIDDEN;
  float s = 0.f;
#pragma unroll
  for (int k = 0; k < HIDDEN; ++k) s = fmaf(hr[k], W2[k * N_CLASSES + c], s);
  z[t] = s;
}

// ------------------------------------------------- layer-2 message passing
__global__ void mp2_kernel(const int* __restrict__ src, const int* __restrict__ dst,
                           const float* __restrict__ dinv,
                           const float* __restrict__ z,
                           float* __restrict__ agg2, int E) {
  int t = blockIdx.x * blockDim.x + threadIdx.x;
  if (t >= E * N_CLASSES) return;
  int e = t / N_CLASSES, c = t - e * N_CLASSES;
  int s = src[e], d = dst[e];
  float w = dinv[s] * dinv[d];
  atomicAdd(&agg2[(size_t)d * N_CLASSES + c], z[(size_t)s * N_CLASSES + c] * w);
}

// self-loop + bias + log_softmax
__global__ void finalize2(const float* __restrict__ agg2, const float* __restrict__ z,
                          const float* __restrict__ dinv,
                          const float* __restrict__ b2,
                          float* __restrict__ out) {
  int i = blockIdx.x * blockDim.x + threadIdx.x;
  if (i >= N_NODES) return;
  float di = dinv[i];
  float sl = di * di;
  float l[N_CLASSES];
  float m = -INFINITY;
#pragma unroll
  for (int c = 0; c < N_CLASSES; ++c) {
    float v = agg2[(size_t)i * N_CLASSES + c] + z[(size_t)i * N_CLASSES + c] * sl + b2[c];
    l[c] = v;
    m = fmaxf(m, v);
  }
  float sum = 0.f;
#pragma unroll
  for (int c = 0; c < N_CLASSES; ++c) sum += expf(l[c] - m);
  float lse = logf(sum);
#pragma unroll
  for (int c = 0; c < N_CLASSES; ++c)
    out[(size_t)i * N_CLASSES + c] = l[c] - m - lse;
}

// ---------------------------------------------------------------- launcher
static inline size_t align256(size_t n) { return (n + 255) & ~(size_t)255; }

extern "C" void kernel_launch(void* const* d_in, const int* in_sizes, int n_in,
                              void* d_out, int out_size, void* d_ws, size_t ws_size,
                              hipStream_t stream) {
  const float* x  = (const float*)d_in[0];
  const int*   ei = (const int*)d_in[1];      // [2, E] int32
  const float* W1 = (const float*)d_in[2];
  const float* b1 = (const float*)d_in[3];
  const float* W2 = (const float*)d_in[4];
  const float* b2 = (const float*)d_in[5];
  float* out = (float*)d_out;

  const int E = in_sizes[1] / 2;
  const int* src = ei;
  const int* dst = ei + E;

  // workspace carve-up (256B aligned)
  char* ws = (char*)d_ws;
  size_t off = 0;
  float* dinv = (float*)(ws + off); off += align256((size_t)N_NODES * 4);
  float* h1   = (float*)(ws + off); off += align256((size_t)N_NODES * HIDDEN * 4);
  float* agg1 = (float*)(ws + off); off += align256((size_t)N_NODES * HIDDEN * 4);
  float* z    = (float*)(ws + off); off += align256((size_t)N_NODES * N_CLASSES * 4);
  float* agg2 = (float*)(ws + off); off += align256((size_t)N_NODES * N_CLASSES * 4);

  const int TB = 256;
  auto blocks = [](long long n, int tb) { return (unsigned)((n + tb - 1) / tb); };

  // 1) init: deg starts at 1.0 (self loop), accumulators zero
  init_f32<<<blocks(N_NODES, TB), TB, 0, stream>>>(dinv, 1.0f, N_NODES);
  init_f32<<<blocks((long long)N_NODES * HIDDEN, TB), TB, 0, stream>>>(
      agg1, 0.0f, N_NODES * HIDDEN);
  init_f32<<<blocks((long long)N_NODES * N_CLASSES, TB), TB, 0, stream>>>(
      agg2, 0.0f, N_NODES * N_CLASSES);

  // 2) symmetric normalization
  degree_kernel<<<blocks(E, TB), TB, 0, stream>>>(dst, E, dinv);
  rsqrt_kernel<<<blocks(N_NODES, TB), TB, 0, stream>>>(dinv, N_NODES);

  // 3) layer 1: dense transform (WMMA), message passing, bias+ReLU
  gemm1_wmma<<<blocks(N_PAIRS, 4), 128, 0, stream>>>(x, W1, h1);
  mp1_kernel<<<blocks((long long)E * HIDDEN, TB), TB, 0, stream>>>(
      src, dst, dinv, h1, agg1, E);
  finalize1<<<blocks((long long)N_NODES * HIDDEN, TB), TB, 0, stream>>>(
      agg1, h1, dinv, b1);

  // 4) layer 2: dense transform, message passing, bias + log_softmax
  gemm2_kernel<<<blocks((long long)N_NODES * N_CLASSES, TB), TB, 0, stream>>>(
      agg1, W2, z);
  mp2_kernel<<<blocks((long long)E * N_CLASSES, TB), TB, 0, stream>>>(
      src, dst, dinv, z, agg2, E);
  finalize2<<<blocks(N_NODES, TB), TB, 0, stream>>>(agg2, z, dinv, b2, out);
}